// blackout2_3599182594543
// MI455X (gfx1250) — compile-verified
//
#include <hip/hip_runtime.h>
#include <hip/hip_bf16.h>

typedef float v2f __attribute__((ext_vector_type(2)));
typedef float v4f __attribute__((ext_vector_type(4)));
typedef float v8f __attribute__((ext_vector_type(8)));

#define TPB     256
#define MAXNCOL 520          // K+1 = 513 for the reference
#define EPSF    1e-10f
#define NEGINF  (-3.402823466e+38f)

// One workgroup per row. Phase 0: gather the 513 sampled logits into LDS.
// Phase A: streaming NT row-max (823 MB total across grid -> HBM bound).
// Phase B: exp/normalize/log over the 513 staged values, deterministic
// LDS tree reductions, per-row partial written to d_ws.
__global__ __launch_bounds__(TPB) void bo2_row_kernel(
    const float* __restrict__ yHat,
    const int*   __restrict__ y,
    const int*   __restrict__ ind,
    float*       __restrict__ partial,
    int C, int K)
{
  __shared__ float vals[MAXNCOL];
  __shared__ float red[TPB];

  const int b = blockIdx.x;
  const int t = threadIdx.x;
  const int ncol = K + 1;
  const float* __restrict__ row = yHat + (size_t)b * (size_t)C;
  const int label = y[b];

  // ---- Phase 0: gather sampled logits (done before NT scan so the scan
  // can be fully non-temporal; no reuse of these lines exists elsewhere).
  for (int idx = t; idx < ncol; idx += TPB) {
    const int col = (idx == 0) ? label : ind[idx - 1];
    vals[idx] = row[col];
  }

  // ---- Phase A: row max. Align to 16B, then stream global_load_b128 NT.
  const int head = (int)(((16u - ((unsigned)(size_t)row & 15u)) & 15u) >> 2);
  const int nvec = (C - head) >> 2;
  const v4f* __restrict__ row4 = (const v4f*)(row + head);

  v4f acc = { NEGINF, NEGINF, NEGINF, NEGINF };
  #pragma unroll 4
  for (int i = t; i < nvec; i += TPB) {
    const v4f v = __builtin_nontemporal_load(&row4[i]);
    acc.x = fmaxf(acc.x, v.x);
    acc.y = fmaxf(acc.y, v.y);
    acc.z = fmaxf(acc.z, v.z);
    acc.w = fmaxf(acc.w, v.w);
  }
  float lmax = fmaxf(fmaxf(acc.x, acc.y), fmaxf(acc.z, acc.w));
  for (int c = t; c < head; c += TPB)                      lmax = fmaxf(lmax, row[c]);
  for (int c = head + (nvec << 2) + t; c < C; c += TPB)    lmax = fmaxf(lmax, row[c]);

  red[t] = lmax;
  __syncthreads();
  for (int off = TPB >> 1; off > 0; off >>= 1) {
    if (t < off) red[t] = fmaxf(red[t], red[t + off]);
    __syncthreads();
  }
  const float mx = red[0];
  __syncthreads();

  // ---- Phase B1: w = exp(v - mx) with label mask; S = sum(w).
  float ps = 0.f;
  for (int idx = t; idx < ncol; idx += TPB) {
    float w = __expf(vals[idx] - mx);
    if (idx > 0 && ind[idx - 1] == label) w = 0.f;   // masked negative
    vals[idx] = w;
    ps += w;
  }
  red[t] = ps;
  __syncthreads();
  for (int off = TPB >> 1; off > 0; off >>= 1) {
    if (t < off) red[t] += red[t + off];
    __syncthreads();
  }
  const float S = red[0];
  __syncthreads();
  const float invS = 1.0f / S;

  // ---- Phase B2: pos = log(out0+eps); neg = sum log(1-out_j+eps).
  float lsum = 0.f;
  for (int idx = t; idx < ncol; idx += TPB) {
    const float o = vals[idx] * invS;
    lsum += (idx == 0) ? __logf(o + EPSF) : __logf(1.0f - o + EPSF);
  }
  red[t] = lsum;
  __syncthreads();
  for (int off = TPB >> 1; off > 0; off >>= 1) {
    if (t < off) red[t] += red[t + off];
    __syncthreads();
  }
  if (t == 0) partial[b] = red[0];
}

// Final reduction: 4096 partials -> scalar. Uses V_WMMA_F32_16X16X4_F32 with
// B = ones as an exact 64-way f32 dot-accumulate (D[m,n] = sum_k A[m,k]).
// Each of the 8 waves folds 512 values via 8 WMMAs; column N=0 of D lives in
// lanes 0 (M=0..7) and 16 (M=8..15) per the ISA C/D layout.
__global__ __launch_bounds__(TPB) void bo2_reduce_kernel(
    const float* __restrict__ partial,
    float*       __restrict__ out,
    int nrows, float inv_n)
{
  __shared__ float wsum[TPB / 32];
  const int t    = threadIdx.x;
  const int wave = t >> 5;
  const int lane = t & 31;
  float wtot = 0.f;
  bool  done = false;

#if __has_builtin(__builtin_amdgcn_wmma_f32_16x16x4_f32)
  if (nrows == 4096) {                    // uniform branch: EXEC stays all-1s
    const v2f ones = { 1.0f, 1.0f };
    v8f c = { 0.f, 0.f, 0.f, 0.f, 0.f, 0.f, 0.f, 0.f };
    const int base = wave * 512 + lane * 2;
    #pragma unroll
    for (int it = 0; it < 8; ++it) {
      v2f a;
      a.x = partial[base + it * 64 + 0];
      a.y = partial[base + it * 64 + 1];
      // 8 args: (neg_a, A, neg_b, B, c_mod, C, reuse_a, reuse_b)
      c = __builtin_amdgcn_wmma_f32_16x16x4_f32(false, a, false, ones,
                                                (short)0, c, false, false);
    }
    float s = c[0] + c[1] + c[2] + c[3] + c[4] + c[5] + c[6] + c[7];
    const float s0  = __shfl(s, 0, 32);   // sum of rows M=0..7  (col 0)
    const float s16 = __shfl(s, 16, 32);  // sum of rows M=8..15 (col 0)
    wtot = s0 + s16;
    done = true;
  }
#endif
  if (!done) {
    float s = 0.f;
    for (int i = t; i < nrows; i += TPB) s += partial[i];
    for (int off = 16; off > 0; off >>= 1) s += __shfl_down(s, off, 32);
    wtot = s;                              // valid at lane 0
  }
  if (lane == 0) wsum[wave] = wtot;
  __syncthreads();
  if (t == 0) {
    float tot = 0.f;
    #pragma unroll
    for (int w = 0; w < TPB / 32; ++w) tot += wsum[w];
    out[0] = -tot * inv_n;                 // loss = -(sum pos + sum neg)/n
  }
}

extern "C" void kernel_launch(void* const* d_in, const int* in_sizes, int n_in,
                              void* d_out, int out_size, void* d_ws, size_t ws_size,
                              hipStream_t stream) {
  const float* yHat = (const float*)d_in[0];   // [B, C] f32
  const int*   y    = (const int*)d_in[1];     // [B]
  const int*   ind  = (const int*)d_in[2];     // [K]
  (void)n_in; (void)out_size; (void)ws_size;

  const int B = in_sizes[1];
  const int C = in_sizes[0] / in_sizes[1];
  const int K = in_sizes[2];

  float* partial = (float*)d_ws;               // B floats of scratch

  bo2_row_kernel<<<B, TPB, 0, stream>>>(yHat, y, ind, partial, C, K);

  const float inv_n = 1.0f / ((float)B * (float)(K + 1));
  bo2_reduce_kernel<<<1, TPB, 0, stream>>>(partial, (float*)d_out, B, inv_n);
}